// DoubleKVCache_27247272526204
// MI455X (gfx1250) — compile-verified
//
#include <hip/hip_runtime.h>

// ---------------------------------------------------------------------------
// DoubleKVCache for MI455X (gfx1250). Pure HBM-bound scatter/copy:
//   ~1.6 GiB total traffic -> ~70us floor @ 23.3 TB/s. No FLOPs, no WMMA.
// CDNA5-specific path used: GLOBAL_LOAD_ASYNC_TO_LDS_B128 /
// GLOBAL_STORE_ASYNC_FROM_LDS_B128 (+ s_wait_asynccnt) for the bulk copy,
// nontemporal B128 everywhere (streaming working set >> 192MB L2).
// ---------------------------------------------------------------------------

#define B_     2
#define H_     32
#define S_MAX_ 8192
#define D_     128
#define S_NEW_ 512
#define BH_    (B_ * H_)
#define CACHE_ELEMS ((size_t)BH_ * S_MAX_ * D_)   // 67,108,864 floats

typedef float f32x4 __attribute__((ext_vector_type(4)));
typedef int   v4i   __attribute__((vector_size(16)));        // matches builtin param

typedef __attribute__((address_space(1))) v4i g_v4i;         // global int4
typedef __attribute__((address_space(3))) v4i l_v4i;         // LDS int4

#if defined(__gfx1250__) && \
    __has_builtin(__builtin_amdgcn_global_load_async_to_lds_b128) && \
    __has_builtin(__builtin_amdgcn_global_store_async_from_lds_b128) && \
    __has_builtin(__builtin_amdgcn_s_wait_asynccnt)
#define HAVE_ASYNC_LDS 1
#else
#define HAVE_ASYNC_LDS 0
#endif

// ---------------------------------------------------------------------------
// Kernel A: bulk copy k_cache -> out1, v_cache -> out2.
// 16,777,216 float4 per tensor. 16384 blocks x 256 threads x 4 float4 each.
// Each lane stages through its own private LDS slots, so only ASYNCcnt waits
// are needed (no barriers): global -> LDS -> global without a VGPR round-trip.
// ---------------------------------------------------------------------------
#if HAVE_ASYNC_LDS
__global__ void __launch_bounds__(256)
bulk_copy_kv_async(const float* __restrict__ kc, const float* __restrict__ vc,
                   float* __restrict__ out1, float* __restrict__ out2) {
    // 8 float4 slots per thread (4 for K, 4 for V): 8*256*16B = 32 KiB LDS.
    __shared__ __align__(16) float stage[8 * 256 * 4];
    const int    tid   = threadIdx.x;
    const size_t base4 = (size_t)blockIdx.x * 1024 + tid;   // float4 units

    #pragma unroll
    for (int j = 0; j < 4; ++j) {
        const size_t idx = (base4 + (size_t)j * 256) * 4;   // float units
        __builtin_amdgcn_global_load_async_to_lds_b128(
            (g_v4i*)((float*)kc + idx),
            (l_v4i*)&stage[((j    ) * 256 + tid) * 4], 0, 0);
        __builtin_amdgcn_global_load_async_to_lds_b128(
            (g_v4i*)((float*)vc + idx),
            (l_v4i*)&stage[((j + 4) * 256 + tid) * 4], 0, 0);
    }
    __builtin_amdgcn_s_wait_asynccnt(0);
    #pragma unroll
    for (int j = 0; j < 4; ++j) {
        const size_t idx = (base4 + (size_t)j * 256) * 4;
        __builtin_amdgcn_global_store_async_from_lds_b128(
            (g_v4i*)(out1 + idx),
            (l_v4i*)&stage[((j    ) * 256 + tid) * 4], 0, 0);
        __builtin_amdgcn_global_store_async_from_lds_b128(
            (g_v4i*)(out2 + idx),
            (l_v4i*)&stage[((j + 4) * 256 + tid) * 4], 0, 0);
    }
    __builtin_amdgcn_s_wait_asynccnt(0);
}
#else
__global__ void __launch_bounds__(256)
bulk_copy_kv(const f32x4* __restrict__ kc, const f32x4* __restrict__ vc,
             f32x4* __restrict__ out1, f32x4* __restrict__ out2) {
    const size_t base = (size_t)blockIdx.x * 1024 + threadIdx.x;
    #pragma unroll
    for (int j = 0; j < 4; ++j) {
        const size_t idx = base + (size_t)j * 256;
        f32x4 a = __builtin_nontemporal_load(&kc[idx]);
        f32x4 b = __builtin_nontemporal_load(&vc[idx]);
        __builtin_nontemporal_store(a, &out1[idx]);
        __builtin_nontemporal_store(b, &out2[idx]);
    }
}
#endif

// ---------------------------------------------------------------------------
// Kernel B: out0[bh][s][d] = kt_cache[bh][d][s]  (128x8192 transpose per bh).
// Tile = 128 s-values x all 128 d. Reads: each wave pulls 512B contiguous
// from one d-row. Writes: each wave emits 512B contiguous for one s-row.
// LDS tile padded to 129 floats/row to spread banks.
// ---------------------------------------------------------------------------
__global__ void __launch_bounds__(256)
transpose_kt(const float* __restrict__ kt, float* __restrict__ out0) {
    __shared__ float tile[128 * 129];                 // 66,048 B
    const int    bh  = blockIdx.x >> 6;               // 64 tiles per bh
    const int    t   = blockIdx.x & 63;
    const int    tid = threadIdx.x;
    const size_t src = (size_t)bh * D_ * S_MAX_ + (size_t)t * 128;  // [d][s0+..]
    const size_t dst = (size_t)bh * S_MAX_ * D_ + (size_t)t * 128 * D_;

    #pragma unroll
    for (int it = 0; it < 16; ++it) {
        const int f = it * 256 + tid;                 // 0..4095
        const int d = f >> 5;                         // 0..127
        const int c = f & 31;                         // float4 column
        f32x4 v = __builtin_nontemporal_load(
            (const f32x4*)(kt + src + (size_t)d * S_MAX_ + 4 * c));
        tile[(4 * c + 0) * 129 + d] = v[0];
        tile[(4 * c + 1) * 129 + d] = v[1];
        tile[(4 * c + 2) * 129 + d] = v[2];
        tile[(4 * c + 3) * 129 + d] = v[3];
    }
    __syncthreads();
    #pragma unroll
    for (int it = 0; it < 16; ++it) {
        const int f  = it * 256 + tid;
        const int s  = f >> 5;                        // 0..127
        const int dc = f & 31;                        // float4 along d
        f32x4 v;
        v[0] = tile[s * 129 + 4 * dc + 0];
        v[1] = tile[s * 129 + 4 * dc + 1];
        v[2] = tile[s * 129 + 4 * dc + 2];
        v[3] = tile[s * 129 + 4 * dc + 3];
        __builtin_nontemporal_store(v, (f32x4*)(out0 + dst + (size_t)s * D_ + 4 * dc));
    }
}

// ---------------------------------------------------------------------------
// Kernel C: scatter new rows. Row (bh,i) of k_val -> out0/out1 at input_pos[i];
// v_val row -> out2. One wave per 512B row, b128 per lane. Runs after A & B.
// ---------------------------------------------------------------------------
__global__ void __launch_bounds__(256)
scatter_new(const int* __restrict__ pos,
            const f32x4* __restrict__ kval, const f32x4* __restrict__ vval,
            f32x4* __restrict__ out0, f32x4* __restrict__ out1,
            f32x4* __restrict__ out2) {
    const int row  = blockIdx.x * 8 + (threadIdx.x >> 5);  // 0..32767
    const int lane = threadIdx.x & 31;
    const int bh   = row >> 9;                              // / S_NEW
    const int i    = row & (S_NEW_ - 1);
    const int p    = pos[i];
    const size_t src = (size_t)row * 32 + lane;
    const size_t dst = ((size_t)bh * S_MAX_ + (size_t)p) * 32 + lane;
    f32x4 k = kval[src];
    f32x4 v = vval[src];
    out0[dst] = k;   // transposed-back view: scattered rows == k_val rows
    out1[dst] = k;
    out2[dst] = v;
}

// ---------------------------------------------------------------------------
extern "C" void kernel_launch(void* const* d_in, const int* in_sizes, int n_in,
                              void* d_out, int out_size, void* d_ws, size_t ws_size,
                              hipStream_t stream) {
    const int*   input_pos = (const int*)  d_in[0];
    const float* k_val     = (const float*)d_in[1];
    const float* v_val     = (const float*)d_in[2];
    const float* k_cache   = (const float*)d_in[3];
    const float* kt_cache  = (const float*)d_in[4];
    const float* v_cache   = (const float*)d_in[5];

    float* out0 = (float*)d_out;               // transposed kt -> [B,H,S,D]
    float* out1 = out0 + CACHE_ELEMS;          // k cache out
    float* out2 = out1 + CACHE_ELEMS;          // v cache out

#if HAVE_ASYNC_LDS
    bulk_copy_kv_async<<<16384, 256, 0, stream>>>(k_cache, v_cache, out1, out2);
#else
    bulk_copy_kv<<<16384, 256, 0, stream>>>((const f32x4*)k_cache, (const f32x4*)v_cache,
                                            (f32x4*)out1, (f32x4*)out2);
#endif
    transpose_kt<<<4096, 256, 0, stream>>>(kt_cache, out0);
    scatter_new<<<4096, 256, 0, stream>>>(input_pos,
                                          (const f32x4*)k_val, (const f32x4*)v_val,
                                          (f32x4*)out0, (f32x4*)out1, (f32x4*)out2);
}